// FFReLU_30339648979369
// MI455X (gfx1250) — compile-verified
//
#include <hip/hip_runtime.h>
#include <stdint.h>

typedef __attribute__((ext_vector_type(2))) float v2f;
typedef __attribute__((ext_vector_type(8))) float v8f;

#define H_DIM    512
#define IN_DIM   13
#define W_STRIDE 7168      // (13+1)*512 floats per weight sample
#define W1_LEN   6656      // H*13
#define R_CNT    256
#define N_CNT    2048
#define LOG_2PI  1.8378770664093453f

// One block (8 wave32) per weight sample r. The block stages the full 28 KB
// weight row in LDS via async global->LDS DMA, then each wave processes 16
// n-tiles of 16 rows with f32 WMMA.
__global__ __launch_bounds__(256) void ffrelu_wmma_kernel(
    const float* __restrict__ x,    // [N, 13]
    const float* __restrict__ y,    // [N, 1]
    const float* __restrict__ w,    // [R, 7168]
    float* __restrict__ out)        // [R, N]
{
    __shared__ float smem_w[W_STRIDE];              // 28672 B << 320 KB LDS

    const int r    = blockIdx.x;
    const int lane = threadIdx.x & 31;
    const int wave = threadIdx.x >> 5;

    const int col = lane & 15;                      // A: M row / B,D: N col
    const int hi  = lane >> 4;                      // half-wave selector

    const float* __restrict__ wr = w + (size_t)r * W_STRIDE;

    // ---- Stage the whole weight row into LDS with async global->LDS copies.
    // 7168 floats = 1792 x 16B chunks; 7 chunks per thread. Tracked on
    // ASYNCcnt; each wave waits its own counter, then the barrier makes the
    // LDS image visible block-wide.
    {
        const uint32_t lds_base = (uint32_t)(uintptr_t)(&smem_w[0]);
        #pragma unroll
        for (int itr = 0; itr < 7; ++itr) {
            const uint32_t byteoff  = (uint32_t)(itr * 256 + threadIdx.x) * 16u;
            const uint32_t lds_addr = lds_base + byteoff;
            asm volatile("global_load_async_to_lds_b128 %0, %1, %2"
                         :: "v"(lds_addr), "v"(byteoff), "s"(wr)
                         : "memory");
        }
        asm volatile("s_wait_asynccnt 0x0" ::: "memory");
    }
    __syncthreads();

    // ---- Each wave handles 16 n-tiles of 16 rows (128 tiles per r). ----
    for (int j = 0; j < 16; ++j) {
        const int n_base = (wave * 16 + j) << 4;

        // A operand: x tile [16 n-rows x K=16 (13 real)], global (L2-hot).
        // 32-bit A 16x4 layout: lane -> M = lane%16; VGPR0/1: K = 2*hi, 2*hi+1.
        v2f a[4];
        {
            const int n = n_base + col;
            #pragma unroll
            for (int c = 0; c < 4; ++c) {
                const int k0 = 4 * c + 2 * hi;
                a[c].x = (k0     < IN_DIM) ? x[n * IN_DIM + k0    ] : 0.0f;
                a[c].y = (k0 + 1 < IN_DIM) ? x[n * IN_DIM + k0 + 1] : 0.0f;
            }
        }

        float mu[8];
        #pragma unroll
        for (int v = 0; v < 8; ++v) mu[v] = 0.0f;

        // Loop over 32 h-tiles of 16 hidden units; B and w2 come from LDS.
        for (int ht = 0; ht < 32; ++ht) {
            const int h = ht * 16 + col;            // this lane's hidden column

            v2f b[4];
            #pragma unroll
            for (int c = 0; c < 4; ++c) {
                const int i0 = 4 * c + 2 * hi;
                b[c].x = (i0     < IN_DIM) ? smem_w[h * IN_DIM + i0    ] : 0.0f;
                b[c].y = (i0 + 1 < IN_DIM) ? smem_w[h * IN_DIM + i0 + 1] : 0.0f;
            }

            v8f d = {};
            #pragma unroll
            for (int c = 0; c < 4; ++c) {
                d = __builtin_amdgcn_wmma_f32_16x16x4_f32(
                        /*neg_a=*/false, a[c],
                        /*neg_b=*/false, b[c],
                        /*c_mod=*/(short)0, d,
                        /*reuse_a=*/false, /*reuse_b=*/false);
            }

            // Fused relu + w2-weighted reduction over this lane's h column.
            // D layout: VGPR v, lane -> (M = v + 8*hi, N = col).
            const float w2v = smem_w[W1_LEN + ht * 16 + col];
            #pragma unroll
            for (int v = 0; v < 8; ++v)
                mu[v] = fmaf(fmaxf(d[v], 0.0f), w2v, mu[v]);
        }

        // Reduce mu across the 16 column-lanes of each half-wave.
        #pragma unroll
        for (int off = 1; off < 16; off <<= 1) {
            #pragma unroll
            for (int v = 0; v < 8; ++v)
                mu[v] += __shfl_xor(mu[v], off, 32);
        }

        // Lanes 0..7 emit rows n_base+0..7 (M=v); lanes 16..23 rows n_base+8..15.
        const int k = lane & 7;
        float msel = mu[0];
        #pragma unroll
        for (int v = 1; v < 8; ++v)
            if (k == v) msel = mu[v];

        if ((lane & 15) < 8) {
            const int row   = n_base + hi * 8 + k;
            const float res = y[row] - msel;
            out[(size_t)r * N_CNT + row] = fmaf(-0.5f * res, res, -0.5f * LOG_2PI);
        }
    }
}

extern "C" void kernel_launch(void* const* d_in, const int* in_sizes, int n_in,
                              void* d_out, int out_size, void* d_ws, size_t ws_size,
                              hipStream_t stream) {
    (void)in_sizes; (void)n_in; (void)d_ws; (void)ws_size; (void)out_size;
    const float* x = (const float*)d_in[0];   // [2048, 13]
    const float* y = (const float*)d_in[1];   // [2048, 1]
    const float* w = (const float*)d_in[2];   // [256, 7168]
    float* out = (float*)d_out;               // [256, 2048]

    // One block per weight sample: 256 blocks x 256 threads (8 wave32).
    dim3 grid(R_CNT), block(256);
    ffrelu_wmma_kernel<<<grid, block, 0, stream>>>(x, y, w, out);
}